// MultiRelationalGCN_197568496059
// MI455X (gfx1250) — compile-verified
//
#include <hip/hip_runtime.h>
#include <math.h>

#define T_DIM 256
#define BB 4
#define MEMC 300
#define NHEADS 3
#define BT (BB * T_DIM)            // 1024 rows (B*T)
#define BTT (BB * T_DIM * T_DIM)   // 262144 per head
#define CMAX 525
#define KSEL 76

typedef float v2f __attribute__((ext_vector_type(2)));
typedef float v8f __attribute__((ext_vector_type(8)));

// ---------------------------------------------------------------------------
// Generic fp32 WMMA GEMM:  C = act( (alpha * A@B + bias_scale*bias) * rowscale + Cin )
// One wave computes a 16x32 tile (two 16x16 accumulators sharing the A frag)
// with V_WMMA_F32_16X16X4_F32.  All hot-loop loads are unconditional: row/col
// indices are clamped to valid addresses (garbage only lands in accumulator
// rows/cols that are never stored); K bounds handled by a zero-padded tail.
// A: M x K (row major, lda).  B: K x N (row major, ldb) or N x K if transB.
// Batched via blockIdx.z with strides sA/sB/sC.
// ---------------------------------------------------------------------------
__global__ void gemm_f32_wmma(const float* __restrict__ A, long long sA,
                              const float* __restrict__ Bm, long long sB,
                              const float* __restrict__ bias, float bias_scale,
                              const float* __restrict__ rowscale,
                              float* __restrict__ C, long long sC,
                              const float* __restrict__ Cin,
                              int M, int N, int K, int lda, int ldb, int ldc,
                              float alpha, int relu, int transB)
{
    int z = blockIdx.z;
    A  += (long long)z * sA;
    Bm += (long long)z * sB;
    C  += (long long)z * sC;
    if (Cin) Cin += (long long)z * sC;

    int tm = blockIdx.y * 16;
    int tn = blockIdx.x * 32;      // two 16-wide N sub-tiles per wave
    int lane = threadIdx.x & 31;
    int lo = lane & 15;
    int hi = lane >> 4;            // 0 or 1

    int am  = tm + lo;             // A-fragment row
    int bn0 = tn + lo;             // B-fragment cols
    int bn1 = tn + 16 + lo;
    // clamp to valid addresses; OOB lanes feed acc entries we never store
    int amc  = am  < M ? am  : M - 1;
    int bn0c = bn0 < N ? bn0 : N - 1;
    int bn1c = bn1 < N ? bn1 : N - 1;

    const float* Ap = A + (long long)amc * lda + hi * 2;   // Ap[k] = A[amc, k + hi*2]

    v8f acc0 = {0.f, 0.f, 0.f, 0.f, 0.f, 0.f, 0.f, 0.f};
    v8f acc1 = {0.f, 0.f, 0.f, 0.f, 0.f, 0.f, 0.f, 0.f};

    int Kmain = K & ~3;
    if (!transB) {
        const float* Bp0 = Bm + (long long)(hi * 2) * ldb + bn0c;
        const float* Bp1 = Bm + (long long)(hi * 2) * ldb + bn1c;
#pragma unroll 2
        for (int k = 0; k < Kmain; k += 4) {
            long long r0 = (long long)k * ldb;
            long long r1 = r0 + ldb;
            v2f a;  a.x  = Ap[k];   a.y  = Ap[k + 1];
            v2f b0; b0.x = Bp0[r0]; b0.y = Bp0[r1];
            v2f b1; b1.x = Bp1[r0]; b1.y = Bp1[r1];
            acc0 = __builtin_amdgcn_wmma_f32_16x16x4_f32(
                false, a, false, b0, (short)0, acc0, false, false);
            acc1 = __builtin_amdgcn_wmma_f32_16x16x4_f32(
                false, a, false, b1, (short)0, acc1, false, false);
        }
    } else {
        const float* Bp0 = Bm + (long long)bn0c * ldb + hi * 2;
        const float* Bp1 = Bm + (long long)bn1c * ldb + hi * 2;
#pragma unroll 2
        for (int k = 0; k < Kmain; k += 4) {
            v2f a;  a.x  = Ap[k];      a.y  = Ap[k + 1];
            v2f b0; b0.x = Bp0[k];     b0.y = Bp0[k + 1];
            v2f b1; b1.x = Bp1[k];     b1.y = Bp1[k + 1];
            acc0 = __builtin_amdgcn_wmma_f32_16x16x4_f32(
                false, a, false, b0, (short)0, acc0, false, false);
            acc1 = __builtin_amdgcn_wmma_f32_16x16x4_f32(
                false, a, false, b1, (short)0, acc1, false, false);
        }
    }
    if (Kmain < K) {               // 1..3 leftover K values: zero-padded tail
        int k0 = Kmain + hi * 2;
        int k1 = k0 + 1;
        v2f a, b0, b1;
        a.x = (k0 < K) ? Ap[Kmain]     : 0.f;
        a.y = (k1 < K) ? Ap[Kmain + 1] : 0.f;
        if (!transB) {
            b0.x = (k0 < K) ? Bm[(long long)k0 * ldb + bn0c] : 0.f;
            b0.y = (k1 < K) ? Bm[(long long)k1 * ldb + bn0c] : 0.f;
            b1.x = (k0 < K) ? Bm[(long long)k0 * ldb + bn1c] : 0.f;
            b1.y = (k1 < K) ? Bm[(long long)k1 * ldb + bn1c] : 0.f;
        } else {
            b0.x = (k0 < K) ? Bm[(long long)bn0c * ldb + k0] : 0.f;
            b0.y = (k1 < K) ? Bm[(long long)bn0c * ldb + k1] : 0.f;
            b1.x = (k0 < K) ? Bm[(long long)bn1c * ldb + k0] : 0.f;
            b1.y = (k1 < K) ? Bm[(long long)bn1c * ldb + k1] : 0.f;
        }
        acc0 = __builtin_amdgcn_wmma_f32_16x16x4_f32(
            false, a, false, b0, (short)0, acc0, false, false);
        acc1 = __builtin_amdgcn_wmma_f32_16x16x4_f32(
            false, a, false, b1, (short)0, acc1, false, false);
    }

    // C/D layout: vgpr r -> row tm + r + 8*hi, col = sub-tile base + lo
    int n0 = tn + lo;
    int n1 = tn + 16 + lo;
    float bv0 = (bias && n0 < N) ? bias_scale * bias[n0] : 0.f;
    float bv1 = (bias && n1 < N) ? bias_scale * bias[n1] : 0.f;
#pragma unroll
    for (int r = 0; r < 8; ++r) {
        int m = tm + r + 8 * hi;
        if (m < M) {
            float rs = rowscale ? rowscale[m] : 1.f;
            if (n0 < N) {
                float v = (alpha * acc0[r] + bv0) * rs;
                if (Cin) v += Cin[(long long)m * ldc + n0];
                if (relu) v = fmaxf(v, 0.f);
                C[(long long)m * ldc + n0] = v;
            }
            if (n1 < N) {
                float v = (alpha * acc1[r] + bv1) * rs;
                if (Cin) v += Cin[(long long)m * ldc + n1];
                if (relu) v = fmaxf(v, 0.f);
                C[(long long)m * ldc + n1] = v;
            }
        }
    }
}

// ---------------------------------------------------------------------------
// Small fused VALU kernels
// ---------------------------------------------------------------------------
__global__ void k_init_mask(const int* __restrict__ ids, int* __restrict__ mask, int n)
{
    int i = blockIdx.x * blockDim.x + threadIdx.x;
    if (i < n) mask[i] = (ids[i] == 0) ? 1 : 0;
}

// channel maxpool (window 5, pad 2) + concat -> xnew (BT x 600)
__global__ void k_build_xnew(const float* __restrict__ x, float* __restrict__ xnew)
{
    int idx = blockIdx.x * blockDim.x + threadIdx.x;
    if (idx >= BT * MEMC) return;
    int m = idx / MEMC, c = idx % MEMC;
    const float* r = x + (long long)m * MEMC;
    int clo = c - 2 < 0 ? 0 : c - 2;
    int chi = c + 2 > MEMC - 1 ? MEMC - 1 : c + 2;
    float mx = r[clo];
    for (int cc = clo + 1; cc <= chi; ++cc) mx = fmaxf(mx, r[cc]);
    xnew[(long long)m * 600 + c] = mx;
    xnew[(long long)m * 600 + 300 + c] = r[c];
}

// adj[h][b,i,j] = softmax_j( relu( sum_c |subj[b,j,c]-obj[b,i,c]| * Wtr[c,h] + btr[h] ) )
// One block per (b,i); subj tiled through LDS (stride 53 avoids bank conflicts).
#define GA_CH 50
__global__ void k_graph_adj(const float* __restrict__ subj, const float* __restrict__ obj,
                            const float* __restrict__ trw, const float* __restrict__ trb,
                            float* __restrict__ adj, float* __restrict__ invd)
{
    __shared__ float obj_s[MEMC];
    __shared__ float trw_s[MEMC * 3];
    __shared__ float sub_s[T_DIM * 53];
    __shared__ float red[T_DIM];

    int b = blockIdx.x / T_DIM, i = blockIdx.x % T_DIM;
    int t = threadIdx.x;

    for (int c = t; c < MEMC; c += T_DIM)
        obj_s[c] = obj[((long long)(b * T_DIM + i)) * MEMC + c];
    for (int c = t; c < MEMC * 3; c += T_DIM) trw_s[c] = trw[c];

    float r0 = trb[0], r1 = trb[1], r2 = trb[2];
    for (int c0 = 0; c0 < MEMC; c0 += GA_CH) {
        __syncthreads();
        for (int x = t; x < T_DIM * GA_CH; x += T_DIM) {
            int jj = x / GA_CH, cc = x % GA_CH;
            sub_s[jj * 53 + cc] = subj[((long long)(b * T_DIM + jj)) * MEMC + c0 + cc];
        }
        __syncthreads();
        for (int cc = 0; cc < GA_CH; ++cc) {
            float d = fabsf(sub_s[t * 53 + cc] - obj_s[c0 + cc]);
            int c = c0 + cc;
            r0 += d * trw_s[c * 3 + 0];
            r1 += d * trw_s[c * 3 + 1];
            r2 += d * trw_s[c * 3 + 2];
        }
    }
    float rv[3] = {fmaxf(r0, 0.f), fmaxf(r1, 0.f), fmaxf(r2, 0.f)};
    __syncthreads();
    for (int h = 0; h < 3; ++h) {
        float v = rv[h];
        red[t] = v; __syncthreads();
        for (int s = 128; s > 0; s >>= 1) {
            if (t < s) red[t] = fmaxf(red[t], red[t + s]);
            __syncthreads();
        }
        float mx = red[0]; __syncthreads();
        float e = __expf(v - mx);
        red[t] = e; __syncthreads();
        for (int s = 128; s > 0; s >>= 1) {
            if (t < s) red[t] += red[t + s];
            __syncthreads();
        }
        float sum = red[0]; __syncthreads();
        adj[(long long)h * BTT + ((long long)(b * T_DIM + i)) * T_DIM + t] = e / sum;
        if (t == 0) invd[h * BT + b * T_DIM + i] = 0.5f;  // softmax rows sum to 1 -> 1/(1+1)
    }
}

__global__ void k_copy2d(const float* __restrict__ X, int ldx,
                         float* __restrict__ Z, int ldz, int rows, int cols)
{
    int idx = blockIdx.x * blockDim.x + threadIdx.x;
    if (idx >= rows * cols) return;
    int m = idx / cols, c = idx % cols;
    Z[(long long)m * ldz + c] = X[(long long)m * ldx + c];
}

__global__ void k_add2d(const float* __restrict__ X, int ldx,
                        const float* __restrict__ Y, int ldy,
                        float* __restrict__ Z, int ldz, int rows, int cols)
{
    int idx = blockIdx.x * blockDim.x + threadIdx.x;
    if (idx >= rows * cols) return;
    int m = idx / cols, c = idx % cols;
    Z[(long long)m * ldz + c] = X[(long long)m * ldx + c] + Y[(long long)m * ldy + c];
}

__global__ void k_score(const float* __restrict__ X, const float* __restrict__ w,
                        const float* __restrict__ b, float* __restrict__ out, int rows)
{
    int m = blockIdx.x * blockDim.x + threadIdx.x;
    if (m >= rows) return;
    float s = b[0];
    const float* r = X + (long long)m * MEMC;
    for (int c = 0; c < MEMC; ++c) s += r[c] * w[c];
    out[m] = s;
}

// para = sigmoid( p(xi) + p(ui) )  (both use same p)
__global__ void k_para(const float* __restrict__ xi, const float* __restrict__ ui,
                       const float* __restrict__ pw, const float* __restrict__ pb,
                       float* __restrict__ para, int rows)
{
    int m = blockIdx.x * blockDim.x + threadIdx.x;
    if (m >= rows) return;
    float s = 2.f * pb[0];
    const float* a = xi + (long long)m * MEMC;
    const float* u = ui + (long long)m * MEMC;
    for (int c = 0; c < MEMC; ++c) s += (a[c] + u[c]) * pw[c];
    para[m] = 1.f / (1.f + __expf(-s));
}

__global__ void k_mix(const float* __restrict__ xi, const float* __restrict__ ui,
                      const float* __restrict__ para, float* __restrict__ out, int rows)
{
    int idx = blockIdx.x * blockDim.x + threadIdx.x;
    if (idx >= rows * MEMC) return;
    int m = idx / MEMC;
    float p = para[m];
    out[idx] = (1.f - p) * xi[idx] + p * ui[idx];
}

// masked softmax over last axis, in place on adj; blocks (i, b, h)
__global__ void k_attn_softmax(float* __restrict__ adj, const int* __restrict__ mask,
                               float* __restrict__ invd)
{
    __shared__ float red[T_DIM];
    int i = blockIdx.x, b = blockIdx.y, h = blockIdx.z, j = threadIdx.x;
    float* row = adj + (long long)h * BTT + ((long long)(b * T_DIM + i)) * T_DIM;
    float v = row[j];
    if (mask[b * T_DIM + j]) v = -10000.f;
    red[j] = v; __syncthreads();
    for (int s = 128; s > 0; s >>= 1) {
        if (j < s) red[j] = fmaxf(red[j], red[j + s]);
        __syncthreads();
    }
    float mx = red[0]; __syncthreads();
    float e = __expf(v - mx);
    red[j] = e; __syncthreads();
    for (int s = 128; s > 0; s >>= 1) {
        if (j < s) red[j] += red[j + s];
        __syncthreads();
    }
    float sum = red[0];
    row[j] = e / sum;
    if (j == 0) invd[h * BT + b * T_DIM + i] = 0.5f;
}

// rank-based selection of the KSEL smallest scores (ties -> lower index, like top_k(-s))
__global__ void k_topk(const float* __restrict__ sct, int* __restrict__ mask)
{
    __shared__ float sc[T_DIM];
    int b = blockIdx.x, h = blockIdx.y, t = threadIdx.x;
    float v = sct[h * BT + b * T_DIM + t];
    sc[t] = v; __syncthreads();
    int cnt = 0;
    for (int j = 0; j < T_DIM; ++j) {
        float o = sc[j];
        cnt += (o < v) || (o == v && j < t);
    }
    if (cnt < KSEL) mask[b * T_DIM + t] = 1;
}

// ---------------------------------------------------------------------------
// Host side
// ---------------------------------------------------------------------------
static void launch_gemm(hipStream_t stream,
                        const float* A, long long sA, const float* Bm, long long sB,
                        const float* bias, float bias_scale, const float* rowscale,
                        float* C, long long sC, const float* Cin,
                        int M, int N, int K, int lda, int ldb, int ldc,
                        float alpha, int relu, int transB, int batches)
{
    dim3 grid((N + 31) / 32, (M + 15) / 16, batches);
    gemm_f32_wmma<<<grid, dim3(32), 0, stream>>>(A, sA, Bm, sB, bias, bias_scale,
                                                 rowscale, C, sC, Cin,
                                                 M, N, K, lda, ldb, ldc,
                                                 alpha, relu, transB);
}

extern "C" void kernel_launch(void* const* d_in, const int* in_sizes, int n_in,
                              void* d_out, int out_size, void* d_ws, size_t ws_size,
                              hipStream_t stream)
{
    (void)in_sizes; (void)n_in; (void)out_size; (void)ws_size;

    const float* inputs = (const float*)d_in[1];        // (B,T,768)
    const int* input_ids = (const int*)d_in[2];         // (B,T)
    auto W = [&](int i) { return (const float*)d_in[i]; };

    // Parameter index map (setup_inputs insertion order, depth-first)
    // 3,4: linear_W w/b ; 5..10: gg subj/obj/tr ; 11..16 attn.q ; 17..22 attn.k
    const int SUBS[4] = {2, 4, 2, 4};
    int wl_w[4][3][4], wl_b[4][3][4], lin_w[4], lin_b[4], score_w[4], score_b[4];
    int idx = 23;
    for (int L = 0; L < 4; ++L) {
        for (int h = 0; h < NHEADS; ++h)
            for (int j = 0; j < SUBS[L]; ++j) { wl_w[L][h][j] = idx++; wl_b[L][h][j] = idx++; }
        lin_w[L] = idx++; lin_b[L] = idx++; score_w[L] = idx++; score_b[L] = idx++;
    }
    int agg_w = idx++, agg_b = idx++;

    // Workspace layout (floats)
    float* ws = (float*)d_ws;
    size_t off = 0;
    auto alloc = [&](size_t n) { float* p = ws + off; off += n; return p; };
    float* gcn_in   = alloc((size_t)BT * MEMC);
    float* layer_out= alloc((size_t)4 * BT * MEMC);
    float* adjbuf   = alloc((size_t)NHEADS * BTT);
    float* invd     = alloc((size_t)NHEADS * BT);
    float* cat      = alloc((size_t)BT * CMAX);
    float* axs      = alloc((size_t)BT * CMAX);
    float* heads    = alloc((size_t)NHEADS * BT * MEMC);
    float* xnew     = alloc((size_t)BT * 600);
    float* subj     = alloc((size_t)BT * MEMC);
    float* objb     = alloc((size_t)BT * MEMC);
    float* qbuf     = alloc((size_t)BT * MEMC);
    float* kbuf     = alloc((size_t)BT * MEMC);
    float* xibuf    = alloc((size_t)BT * MEMC);
    float* uibuf    = alloc((size_t)BT * MEMC);
    float* parabuf  = alloc((size_t)BT);
    float* sct      = alloc((size_t)NHEADS * BT);
    int*   mask     = (int*)alloc((size_t)BT);

    // 1) embed: gcn_in = inputs @ W(768x300) + b
    launch_gemm(stream, inputs, 0, W(3), 0, W(4), 1.f, nullptr,
                gcn_in, 0, nullptr, BT, MEMC, 768, 768, MEMC, MEMC, 1.f, 0, 0, 1);

    // 2) graph generate
    k_build_xnew<<<(BT * MEMC + 255) / 256, 256, 0, stream>>>(gcn_in, xnew);
    launch_gemm(stream, xnew, 0, W(5), 0, W(6), 1.f, nullptr,
                subj, 0, nullptr, BT, MEMC, 600, 600, MEMC, MEMC, 1.f, 0, 0, 1);
    launch_gemm(stream, xnew, 0, W(7), 0, W(8), 1.f, nullptr,
                objb, 0, nullptr, BT, MEMC, 600, 600, MEMC, MEMC, 1.f, 0, 0, 1);
    k_graph_adj<<<BT, T_DIM, 0, stream>>>(subj, objb, W(9), W(10), adjbuf, invd);
    k_init_mask<<<(BT + 255) / 256, 256, 0, stream>>>(input_ids, mask, BT);

    const float* gin = gcn_in;
    for (int L = 0; L < 4; ++L) {
        int ns = SUBS[L];
        int hd = MEMC / ns;

        if (L >= 2) {
            // attention rebuilds adj + invd from gin and current mask
            for (int s = 0; s < 2; ++s) {
                int base = (s == 0) ? 11 : 17;
                launch_gemm(stream, gin, 0, W(base + 0), 0, W(base + 1), 1.f, nullptr,
                            xibuf, 0, nullptr, BT, MEMC, MEMC, MEMC, MEMC, MEMC, 1.f, 0, 0, 1);
                launch_gemm(stream, gin, 0, W(base + 2), 0, W(base + 3), 1.f, nullptr,
                            uibuf, 0, nullptr, BT, MEMC, MEMC, MEMC, MEMC, MEMC, 1.f, 0, 0, 1);
                k_para<<<(BT + 255) / 256, 256, 0, stream>>>(xibuf, uibuf, W(base + 4),
                                                             W(base + 5), parabuf, BT);
                k_mix<<<(BT * MEMC + 255) / 256, 256, 0, stream>>>(
                    xibuf, uibuf, parabuf, (s == 0 ? qbuf : kbuf), BT);
            }
            for (int h = 0; h < NHEADS; ++h)
                launch_gemm(stream, qbuf + h * 100, (long long)T_DIM * MEMC,
                            kbuf + h * 100, (long long)T_DIM * MEMC,
                            nullptr, 0.f, nullptr,
                            adjbuf + (long long)h * BTT, (long long)T_DIM * T_DIM, nullptr,
                            T_DIM, T_DIM, 100, MEMC, MEMC, T_DIM,
                            0.1f /*1/sqrt(100)*/, 0, 1 /*transB*/, BB);
            k_attn_softmax<<<dim3(T_DIM, BB, NHEADS), T_DIM, 0, stream>>>(adjbuf, mask, invd);
        }

        for (int h = 0; h < NHEADS; ++h) {
            k_copy2d<<<(BT * MEMC + 255) / 256, 256, 0, stream>>>(gin, MEMC, cat, CMAX, BT, MEMC);
            int C = MEMC;
            for (int l = 0; l < ns; ++l) {
                // Ax = adj_h @ cat   (batched over B)
                launch_gemm(stream, adjbuf + (long long)h * BTT, (long long)T_DIM * T_DIM,
                            cat, (long long)T_DIM * CMAX, nullptr, 0.f, nullptr,
                            axs, (long long)T_DIM * CMAX, nullptr,
                            T_DIM, C, T_DIM, T_DIM, CMAX, CMAX, 1.f, 0, 0, BB);
                // S = Ax + cat
                k_add2d<<<(BT * C + 255) / 256, 256, 0, stream>>>(axs, CMAX, cat, CMAX,
                                                                  axs, CMAX, BT, C);
                // g = relu((S@W + 2b) / denom) -> appended into cat
                launch_gemm(stream, axs, 0, W(wl_w[L][h][l]), 0, W(wl_b[L][h][l]), 2.f,
                            invd + h * BT,
                            cat + MEMC + l * hd, 0, nullptr,
                            BT, hd, C, CMAX, hd, CMAX, 1.f, 1 /*relu*/, 0, 1);
                C += hd;
            }
            // heads_out[h] = concat(g's) + gin
            k_add2d<<<(BT * MEMC + 255) / 256, 256, 0, stream>>>(
                cat + MEMC, CMAX, gin, MEMC, heads + (long long)h * BT * MEMC, MEMC, BT, MEMC);
            k_score<<<(BT + 255) / 256, 256, 0, stream>>>(
                heads + (long long)h * BT * MEMC, W(score_w[L]), W(score_b[L]),
                sct + h * BT, BT);
        }

        // out = concat(heads) @ lin  -> 3 accumulating GEMMs
        float* lout = layer_out + (long long)L * BT * MEMC;
        for (int h = 0; h < NHEADS; ++h)
            launch_gemm(stream, heads + (long long)h * BT * MEMC, 0,
                        W(lin_w[L]) + (long long)h * MEMC * MEMC, 0,
                        (h == 0 ? W(lin_b[L]) : nullptr), 1.f, nullptr,
                        lout, 0, (h == 0 ? nullptr : lout),
                        BT, MEMC, MEMC, MEMC, MEMC, MEMC, 1.f, 0, 0, 1);

        if (L < 3)  // layer 3's mask output is never consumed
            k_topk<<<dim3(BB, NHEADS), T_DIM, 0, stream>>>(sct, mask);

        gin = lout;
    }

    // final agg: out = concat(layer_list) @ Wagg + b  -> 4 accumulating GEMMs
    float* outp = (float*)d_out;
    for (int j = 0; j < 4; ++j)
        launch_gemm(stream, layer_out + (long long)j * BT * MEMC, 0,
                    W(agg_w) + (long long)j * MEMC * MEMC, 0,
                    (j == 0 ? W(agg_b) : nullptr), 1.f, nullptr,
                    outp, 0, (j == 0 ? nullptr : outp),
                    BT, MEMC, MEMC, MEMC, MEMC, MEMC, 1.f, 0, 0, 1);
}